// FeatureLearningNet_39367670236039
// MI455X (gfx1250) — compile-verified
//
#include <hip/hip_runtime.h>
#include <hip/hip_bf16.h>

typedef __attribute__((ext_vector_type(2))) float v2f;
typedef __attribute__((ext_vector_type(4))) float v4f;
typedef __attribute__((ext_vector_type(8))) float v8f;

#define KVOX 15000
#define VP 35
#define MT 48                    // padded M rows (3 tiles of 16)
#define NPAD (KVOX * (MT - VP))  // pad rows counted into raw BN stats
#define SLOTS 256                // stats slot partitioning (atomic contention)

// ---------------------------------------------------------------- zeroing ---
__global__ void k_zero4(v4f* __restrict__ p, long n4) {
  long i = (long)blockIdx.x * blockDim.x + threadIdx.x;
  if (i < n4) {
    v4f z = {0.f, 0.f, 0.f, 0.f};
    __builtin_nontemporal_store(z, p + i);
  }
}

// ------------------------------------------------------- stage 1: 7->16 -----
__global__ void __launch_bounds__(128)
k_vfe1(const float* __restrict__ feat, const float* __restrict__ w1,
       const float* __restrict__ b1,
       float* __restrict__ pw1, float* __restrict__ maskb,
       float* __restrict__ stats1)
{
  __shared__ __align__(16) float sA[MT * 8];   // feature tile, K padded 7->8
  __shared__ __align__(16) float sW[8 * 16];   // w1 padded
  __shared__ float sB[16];
  __shared__ float sSum[16];
  __shared__ float sSq[16];

  const int k = blockIdx.x;
  const int t = threadIdx.x;

  for (int i = t; i < MT * 8; i += 128) sA[i] = 0.f;
  if (t < 16) { sB[t] = b1[t]; sSum[t] = 0.f; sSq[t] = 0.f; }
  sW[t] = (t < 7 * 16) ? w1[t] : 0.f;          // 128 threads cover [8][16]
  __syncthreads();
  for (int i = t; i < VP * 7; i += 128) {
    int p = i / 7, c = i % 7;
    sA[p * 8 + c] = feat[((long)k * VP + p) * 7 + c];
  }
  __syncthreads();

  if (t < VP) {
    float mx = sA[t * 8];
    for (int c = 1; c < 7; ++c) mx = fmaxf(mx, sA[t * 8 + c]);
    maskb[(long)k * VP + t] = (mx != 0.f) ? 1.f : 0.f;
  }

  const int w = t >> 5, lane = t & 31, l = lane & 15, hi = lane >> 4;
  if (w < 3) {                                 // 3 M-tiles, one N-tile
    const int mbase = w * 16;
    v8f acc = {0.f, 0.f, 0.f, 0.f, 0.f, 0.f, 0.f, 0.f};
#pragma unroll
    for (int kq = 0; kq < 2; ++kq) {
      const int k0 = kq * 4;
      v2f a = *(const v2f*)&sA[(mbase + l) * 8 + k0 + 2 * hi];
      v2f b;
      b.x = sW[(k0 + 2 * hi + 0) * 16 + l];
      b.y = sW[(k0 + 2 * hi + 1) * 16 + l];
      acc = __builtin_amdgcn_wmma_f32_16x16x4_f32(false, a, false, b,
                                                  (short)0, acc, false, false);
    }
    const float bias = sB[l];
    float lsum = 0.f, lsq = 0.f;
#pragma unroll
    for (int r = 0; r < 8; ++r) {              // unconditional: padded layout
      int row = mbase + r + 8 * hi;
      float v = fmaxf(acc[r] + bias, 0.f);
      pw1[((long)k * MT + row) * 16 + l] = v;
      lsum += v; lsq += v * v;                 // pads subtracted in finalize
    }
    lsum += __shfl_xor(lsum, 16);
    lsq  += __shfl_xor(lsq, 16);
    if (hi == 0) { atomicAdd(&sSum[l], lsum); atomicAdd(&sSq[l], lsq); }
  }
  __syncthreads();
  if (t < 16) {
    int slot = k & (SLOTS - 1);
    atomicAdd(&stats1[slot * 32 + t], sSum[t]);
    atomicAdd(&stats1[slot * 32 + 16 + t], sSq[t]);
  }
}

// -------------------------------------------------- BN finalize (both) ------
// Raw sums include NPAD pad rows whose value is exactly relu(bias[c])
// (zero input rows) -> subtract analytically, then biased mean/var.
__global__ void k_fin(const float* __restrict__ stats,
                      const float* __restrict__ g, const float* __restrict__ be,
                      const float* __restrict__ bias,
                      float* __restrict__ sc, float* __restrict__ sh, int nch)
{
  int c = threadIdx.x;
  if (c >= nch) return;
  float s = 0.f, q = 0.f;
  for (int i = 0; i < SLOTS; ++i) {
    s += stats[i * 2 * nch + c];
    q += stats[i * 2 * nch + nch + c];
  }
  float pv = fmaxf(bias[c], 0.f);
  s -= (float)NPAD * pv;
  q -= (float)NPAD * pv * pv;
  const float invN = 1.0f / (float)(KVOX * VP);
  float mean = s * invN;
  float var  = q * invN - mean * mean;
  float scale = g[c] * rsqrtf(var + 1e-5f);
  sc[c] = scale;
  sh[c] = be[c] - mean * scale;
}

// ------------------------------------- stage 2: 32->64 (stats / final) ------
template <bool FINAL>
__global__ void __launch_bounds__(128)
k_vfe2(const float* __restrict__ pw1, const float* __restrict__ maskb,
       const int* __restrict__ coord,
       const float* __restrict__ sc1, const float* __restrict__ sh1,
       const float* __restrict__ w2, const float* __restrict__ b2,
       const float* __restrict__ sc2, const float* __restrict__ sh2,
       float* __restrict__ stats2, float* __restrict__ out)
{
  constexpr int XS = 34;       // x2 LDS stride (even -> 8B aligned float2)
  constexpr int OS = 66;       // pw2n LDS stride
  __shared__ __align__(16) float sP[VP * 16];
  __shared__ __align__(16) float sX[MT * XS];
  __shared__ __align__(16) float sW2[32 * 64];
  __shared__ float sB2[64];
  __shared__ float sM[VP];
  __shared__ float sAgg[16];
  __shared__ float sSum[64];
  __shared__ float sSq[64];
  __shared__ __align__(16) float sO[FINAL ? MT * OS : 4];

  const int k = blockIdx.x;
  const int t = threadIdx.x;

  for (int i = t; i < MT * XS; i += 128) sX[i] = 0.f;
  for (int i = t; i < 512; i += 128)                       // w2 as float4
    ((v4f*)sW2)[i] = ((const v4f*)w2)[i];
  if (t < 64) { sB2[t] = b2[t]; sSum[t] = 0.f; sSq[t] = 0.f; }
  {
    const v4f* src = (const v4f*)(pw1 + (long)k * MT * 16);
    for (int i = t; i < VP * 16 / 4; i += 128) {           // BN1 on load
      v4f v = src[i];
      int c0 = (4 * i) & 15;                               // 16 % 4 == 0
      v4f r = {v.x * sc1[c0 + 0] + sh1[c0 + 0],
               v.y * sc1[c0 + 1] + sh1[c0 + 1],
               v.z * sc1[c0 + 2] + sh1[c0 + 2],
               v.w * sc1[c0 + 3] + sh1[c0 + 3]};
      ((v4f*)sP)[i] = r;
    }
  }
  if (t < VP) sM[t] = maskb[(long)k * VP + t];
  __syncthreads();

  if (t < 16) {                                            // voxel max (pre-mask)
    float mx = sP[t];
    for (int p = 1; p < VP; ++p) mx = fmaxf(mx, sP[p * 16 + t]);
    sAgg[t] = mx;
  }
  __syncthreads();
  for (int i = t; i < VP * 32; i += 128) {                 // x2 = concat * mask
    int p = i >> 5, c = i & 31;
    float v = (c < 16) ? sP[p * 16 + c] : sAgg[c - 16];
    sX[p * XS + c] = v * sM[p];
  }
  __syncthreads();

  const int w = t >> 5, lane = t & 31, l = lane & 15, hi = lane >> 4;
  const int nb = w * 16;                       // wave w owns 16 output cols
  const float bias = sB2[nb + l];
  float scl = 0.f, shf = 0.f;
  if (FINAL) { scl = sc2[nb + l]; shf = sh2[nb + l]; }

  v8f acc[3];
#pragma unroll
  for (int m = 0; m < 3; ++m)
    acc[m] = (v8f){0.f, 0.f, 0.f, 0.f, 0.f, 0.f, 0.f, 0.f};
#pragma unroll
  for (int kq = 0; kq < 8; ++kq) {             // B loaded once per k-step
    const int k0 = kq * 4;
    v2f b;
    b.x = sW2[(k0 + 2 * hi + 0) * 64 + nb + l];
    b.y = sW2[(k0 + 2 * hi + 1) * 64 + nb + l];
#pragma unroll
    for (int m = 0; m < 3; ++m) {
      v2f a = *(const v2f*)&sX[(m * 16 + l) * XS + k0 + 2 * hi];
      acc[m] = __builtin_amdgcn_wmma_f32_16x16x4_f32(false, a, false, b,
                                                     (short)0, acc[m],
                                                     false, false);
    }
  }

  float lsum = 0.f, lsq = 0.f;
#pragma unroll
  for (int m = 0; m < 3; ++m) {
#pragma unroll
    for (int r = 0; r < 8; ++r) {              // unconditional epilogue
      int row = m * 16 + r + 8 * hi;
      float v = fmaxf(acc[m][r] + bias, 0.f);
      if (FINAL) {
        sO[row * OS + nb + l] = v * scl + shf; // pad rows written, never read
      } else {
        lsum += v; lsq += v * v;               // pads subtracted in finalize
      }
    }
  }

  if (!FINAL) {
    lsum += __shfl_xor(lsum, 16);
    lsq  += __shfl_xor(lsq, 16);
    if (hi == 0) { sSum[nb + l] = lsum; sSq[nb + l] = lsq; }
  }
  __syncthreads();

  if (!FINAL) {
    if (t < 64) {
      int slot = k & (SLOTS - 1);
      atomicAdd(&stats2[slot * 128 + t], sSum[t]);
      atomicAdd(&stats2[slot * 128 + 64 + t], sSq[t]);
    }
  } else {
    float vox;
    if (t < 64) {
      float m = -3.4e38f;
      for (int p = 0; p < VP; ++p) m = fmaxf(m, sO[p * OS + t] * sM[p]);
      vox = m;
    } else {
      int ch = t - 64;
      float agg = -3.4e38f;
      for (int p = 0; p < VP; ++p) agg = fmaxf(agg, sO[p * OS + ch]);
      float m = -3.4e38f;
      for (int p = 0; p < VP; ++p) m = fmaxf(m, agg * sM[p]);
      vox = m;
    }
    const int* cd = coord + (long)k * 4;
    long idx = ((((long)cd[0] * 10 + cd[1]) * 400 + cd[2]) * 352 + cd[3]) * 128 + t;
    atomicAdd(out + idx, vox);                 // duplicate coords must sum
  }
}

// ---------------------------------------------------------------- driver ----
extern "C" void kernel_launch(void* const* d_in, const int* in_sizes, int n_in,
                              void* d_out, int out_size, void* d_ws, size_t ws_size,
                              hipStream_t stream) {
  const float* feat  = (const float*)d_in[0];
  const int*   coord = (const int*)  d_in[1];
  // d_in[2] = bs (==1, unused)
  const float* w1  = (const float*)d_in[3];
  const float* b1  = (const float*)d_in[4];
  const float* g1  = (const float*)d_in[5];
  const float* be1 = (const float*)d_in[6];
  const float* w2  = (const float*)d_in[7];
  const float* b2  = (const float*)d_in[8];
  const float* g2  = (const float*)d_in[9];
  const float* be2 = (const float*)d_in[10];
  float* out = (float*)d_out;

  float* ws     = (float*)d_ws;
  float* pw1    = ws;                               // K*MT*16 = 11.52M floats
  float* maskb  = pw1 + (long)KVOX * MT * 16;       // K*VP    = 525k
  float* stats1 = maskb + (long)KVOX * VP;          // SLOTS*32
  float* sc1    = stats1 + SLOTS * 32;              // 16
  float* sh1    = sc1 + 16;                         // 16
  float* stats2 = sh1 + 16;                         // SLOTS*128
  float* sc2    = stats2 + SLOTS * 128;             // 64
  float* sh2    = sc2 + 64;                         // 64

  // 1) zero the 721MB dense output (dominant memory cost) + stats scratch
  long n4 = (long)out_size / 4;
  k_zero4<<<(int)((n4 + 255) / 256), 256, 0, stream>>>((v4f*)out, n4);
  long nstats = SLOTS * 32 + 16 + 16 + SLOTS * 128 + 64 + 64;  // %4 == 0
  k_zero4<<<(int)((nstats / 4 + 255) / 256), 256, 0, stream>>>((v4f*)stats1,
                                                               nstats / 4);
  // 2) VFE-1 GEMM + ReLU + BN partial stats
  k_vfe1<<<KVOX, 128, 0, stream>>>(feat, w1, b1, pw1, maskb, stats1);
  k_fin<<<1, 32, 0, stream>>>(stats1, g1, be1, b1, sc1, sh1, 16);
  // 3) VFE-2 GEMM (stats pass, no pw2 stored)
  k_vfe2<false><<<KVOX, 128, 0, stream>>>(pw1, maskb, coord, sc1, sh1,
                                          w2, b2, nullptr, nullptr,
                                          stats2, nullptr);
  k_fin<<<1, 64, 0, stream>>>(stats2, g2, be2, b2, sc2, sh2, 64);
  // 4) VFE-2 GEMM recompute + BN2 + voxel max + scatter-add
  k_vfe2<true><<<KVOX, 128, 0, stream>>>(pw1, maskb, coord, sc1, sh1,
                                         w2, b2, sc2, sh2,
                                         nullptr, out);
}